// RelateModule_26001732010249
// MI455X (gfx1250) — compile-verified
//
#include <hip/hip_runtime.h>

#define Nn 1024
#define Ee 32
#define Ll 8

typedef __attribute__((ext_vector_type(2))) float v2f;
typedef __attribute__((ext_vector_type(8))) float v8f;

// ---------------------------------------------------------------------------
// 1) query[e] = dot(c_i, w_c[e,:]) + b_c[e]   (32 outputs, one wave)
// ---------------------------------------------------------------------------
__global__ __launch_bounds__(32) void query_kernel(const float* __restrict__ c_i,
                                                   const float* __restrict__ w_c,
                                                   const float* __restrict__ b_c,
                                                   float* __restrict__ query) {
    int e = threadIdx.x;
    const float4* c4 = reinterpret_cast<const float4*>(c_i);
    const float4* w4 = reinterpret_cast<const float4*>(w_c + (size_t)e * 1024);
    float acc = 0.f;
    for (int i = 0; i < 256; ++i) {
        float4 c = c4[i], w = w4[i];
        acc += c.x * w.x + c.y * w.y + c.z * w.z + c.w * w.w;
    }
    query[e] = acc + b_c[e];
}

// ---------------------------------------------------------------------------
// 2) wv[m,y] = mono[m,y] * sum_e cm[e] * ( relu(I+fe) - (I-fe)^2 )
//    I = query (m==0) else feat[m-1,y,:]
// ---------------------------------------------------------------------------
__global__ __launch_bounds__(256) void wv_kernel(const float* __restrict__ query,
                                                 const float* __restrict__ feat,
                                                 const float* __restrict__ feat_edge,
                                                 const float* __restrict__ mono,
                                                 const float* __restrict__ cm,
                                                 float* __restrict__ Wv) {
    int y = blockIdx.x * 256 + threadIdx.x;
    int m = blockIdx.y;
    const float* Irow = (m == 0) ? query : (feat + ((size_t)(m - 1) * Nn + y) * Ee);
    const float* fer  = feat_edge + (size_t)y * Ee;
    float acc = 0.f;
#pragma unroll
    for (int e4 = 0; e4 < 8; ++e4) {
        float4 iv = *reinterpret_cast<const float4*>(Irow + e4 * 4);
        float4 fv = *reinterpret_cast<const float4*>(fer  + e4 * 4);
        float4 cv = *reinterpret_cast<const float4*>(cm   + e4 * 4);
        float d, s;
        d = iv.x - fv.x; s = iv.x + fv.x; acc += cv.x * (fmaxf(s, 0.f) - d * d);
        d = iv.y - fv.y; s = iv.y + fv.y; acc += cv.y * (fmaxf(s, 0.f) - d * d);
        d = iv.z - fv.z; s = iv.z + fv.z; acc += cv.z * (fmaxf(s, 0.f) - d * d);
        d = iv.w - fv.w; s = iv.w + fv.w; acc += cv.w * (fmaxf(s, 0.f) - d * d);
    }
    size_t o = (size_t)m * Nn + y;
    Wv[o] = mono[o] * acc;
}

// ---------------------------------------------------------------------------
// 3) Sxy[x,y] = sum_l att_stack[x,y,l] * stack_ptr[l]
// ---------------------------------------------------------------------------
__global__ __launch_bounds__(256) void s_kernel(const float* __restrict__ att,
                                                const float* __restrict__ sp,
                                                float* __restrict__ Sxy) {
    int idx = blockIdx.x * 256 + threadIdx.x;  // over N*N
    float4 p0 = *reinterpret_cast<const float4*>(sp);
    float4 p1 = *reinterpret_cast<const float4*>(sp + 4);
    float4 a0 = *reinterpret_cast<const float4*>(att + (size_t)idx * 8);
    float4 a1 = *reinterpret_cast<const float4*>(att + (size_t)idx * 8 + 4);
    Sxy[idx] = a0.x * p0.x + a0.y * p0.y + a0.z * p0.z + a0.w * p0.w +
               a1.x * p1.x + a1.y * p1.y + a1.z * p1.z + a1.w * p1.w;
}

// ---------------------------------------------------------------------------
// 4) C[m,x] = sum_y Wv[m,y] * Sxy[x,y]   (1024^3 fp32 GEMM via WMMA f32 16x16x4)
//    Block tile 64x64, BK=16, 8 waves (4 in M x 2 in N), 2 accum tiles/wave.
// ---------------------------------------------------------------------------
__global__ __launch_bounds__(256) void gemm_kernel(const float* __restrict__ W,
                                                   const float* __restrict__ Sxy,
                                                   float* __restrict__ C) {
    __shared__ float A_lds[64][17];   // [m][k], padded
    __shared__ float B_lds[16][65];   // [k][n], padded (transposed on load)

    const int tid  = threadIdx.x;
    const int lane = tid & 31;
    const int wave = tid >> 5;
    const int hlf  = lane >> 4;       // 0: lanes 0-15, 1: lanes 16-31
    const int ml   = lane & 15;
    const int wm   = (wave & 3) * 16; // wave M offset in block tile
    const int wn   = (wave >> 2) * 32;// wave N offset in block tile
    const int m0   = blockIdx.y * 64;
    const int x0   = blockIdx.x * 64;
    const int lr   = tid >> 2;        // 0..63 : row index for tile loads
    const int lq   = tid & 3;         // quad of 4 floats

    v8f c0 = {};
    v8f c1 = {};

    for (int k0 = 0; k0 < Nn; k0 += 16) {
        // A tile: 64 rows of W, 16 k's (row-major, coalesced float4)
        float4 a4 = *reinterpret_cast<const float4*>(W + (size_t)(m0 + lr) * Nn + k0 + lq * 4);
        A_lds[lr][lq * 4 + 0] = a4.x;
        A_lds[lr][lq * 4 + 1] = a4.y;
        A_lds[lr][lq * 4 + 2] = a4.z;
        A_lds[lr][lq * 4 + 3] = a4.w;
        // B tile: B[k][n] = Sxy[x0+n][k0+k] (read rows of Sxy, transpose into LDS)
        float4 b4 = *reinterpret_cast<const float4*>(Sxy + (size_t)(x0 + lr) * Nn + k0 + lq * 4);
        B_lds[lq * 4 + 0][lr] = b4.x;
        B_lds[lq * 4 + 1][lr] = b4.y;
        B_lds[lq * 4 + 2][lr] = b4.z;
        B_lds[lq * 4 + 3][lr] = b4.w;
        __syncthreads();

        if (k0 + 16 < Nn) {
            __builtin_prefetch(W   + (size_t)(m0 + lr) * Nn + k0 + 16, 0, 1);
            __builtin_prefetch(Sxy + (size_t)(x0 + lr) * Nn + k0 + 16, 0, 1);
        }

#pragma unroll
        for (int kk = 0; kk < 4; ++kk) {
            // 16x4 fp32 A fragment: lanes 0-15 hold K0/K1, lanes 16-31 hold K2/K3
            const int ka = kk * 4 + hlf * 2;
            v2f a;
            a.x = A_lds[wm + ml][ka];
            a.y = A_lds[wm + ml][ka + 1];
            v2f b0, b1;
            b0.x = B_lds[ka][wn + ml];
            b0.y = B_lds[ka + 1][wn + ml];
            b1.x = B_lds[ka][wn + 16 + ml];
            b1.y = B_lds[ka + 1][wn + 16 + ml];
            c0 = __builtin_amdgcn_wmma_f32_16x16x4_f32(false, a, false, b0, (short)0, c0, false, false);
            c1 = __builtin_amdgcn_wmma_f32_16x16x4_f32(false, a, false, b1, (short)0, c1, false, false);
        }
        __syncthreads();
    }

    // C/D layout: VGPR r -> (M=r, N=lane) lanes 0-15 ; (M=8+r, N=lane-16) lanes 16-31
#pragma unroll
    for (int r = 0; r < 8; ++r) {
        int row = m0 + wm + hlf * 8 + r;
        C[(size_t)row * Nn + x0 + wn + ml]      = c0[r];
        C[(size_t)row * Nn + x0 + wn + 16 + ml] = c1[r];
    }
}

// ---------------------------------------------------------------------------
// 5) norm[m] = max_x C[m,x]; if <= 1 -> 1
// ---------------------------------------------------------------------------
__global__ __launch_bounds__(256) void norm_kernel(const float* __restrict__ C,
                                                   float* __restrict__ norm) {
    __shared__ float red[256];
    int m = blockIdx.x;
    const float* row = C + (size_t)m * Nn;
    float v = -1e38f;
    for (int i = threadIdx.x; i < Nn; i += 256) v = fmaxf(v, row[i]);
    red[threadIdx.x] = v;
    __syncthreads();
    for (int s = 128; s > 0; s >>= 1) {
        if (threadIdx.x < s) red[threadIdx.x] = fmaxf(red[threadIdx.x], red[threadIdx.x + s]);
        __syncthreads();
    }
    if (threadIdx.x == 0) {
        float nm = red[0];
        norm[m] = (nm <= 1.0f) ? 1.0f : nm;
    }
}

// ---------------------------------------------------------------------------
// 6) att_stack_new[x,m,l] = (C[m,x]/norm[m])*ptr[l] + att_stack[x,m,l]*(1-ptr[l])
//    32x32 (m,x) tile, LDS transpose so both C reads and float4 blends coalesce.
// ---------------------------------------------------------------------------
__global__ __launch_bounds__(256) void final_kernel(const float* __restrict__ C,
                                                    const float* __restrict__ norm,
                                                    const float* __restrict__ att,
                                                    const float* __restrict__ sp,
                                                    float* __restrict__ out) {
    __shared__ float Ct[32][33];      // [x_local][m_local]
    __shared__ float s_norm[32];
    const int tid = threadIdx.x;
    const int m0 = blockIdx.x * 32;
    const int x0 = blockIdx.y * 32;
    const int col = tid & 31;         // x_local during load
    const int rb  = tid >> 5;         // 0..7
#pragma unroll
    for (int i = 0; i < 4; ++i) {
        int mlc = rb + i * 8;
        Ct[col][mlc] = C[(size_t)(m0 + mlc) * Nn + x0 + col];
    }
    if (tid < 32) s_norm[tid] = norm[m0 + tid];
    __syncthreads();

    float4 p0 = *reinterpret_cast<const float4*>(sp);
    float4 p1 = *reinterpret_cast<const float4*>(sp + 4);
    float4 q0 = make_float4(1.f - p0.x, 1.f - p0.y, 1.f - p0.z, 1.f - p0.w);
    float4 q1 = make_float4(1.f - p1.x, 1.f - p1.y, 1.f - p1.z, 1.f - p1.w);

#pragma unroll
    for (int j = 0; j < 4; ++j) {
        int idx = tid + j * 256;
        int xl  = idx >> 5;
        int mlc = idx & 31;
        float v = Ct[xl][mlc] / s_norm[mlc];
        size_t base = (((size_t)(x0 + xl)) * Nn + (m0 + mlc)) * Ll;
        float4 a0 = *reinterpret_cast<const float4*>(att + base);
        float4 a1 = *reinterpret_cast<const float4*>(att + base + 4);
        float4 o0, o1;
        o0.x = v * p0.x + a0.x * q0.x;
        o0.y = v * p0.y + a0.y * q0.y;
        o0.z = v * p0.z + a0.z * q0.z;
        o0.w = v * p0.w + a0.w * q0.w;
        o1.x = v * p1.x + a1.x * q1.x;
        o1.y = v * p1.y + a1.y * q1.y;
        o1.z = v * p1.z + a1.z * q1.z;
        o1.w = v * p1.w + a1.w * q1.w;
        *reinterpret_cast<float4*>(out + base)     = o0;
        *reinterpret_cast<float4*>(out + base + 4) = o1;
    }
}

// ---------------------------------------------------------------------------
// 7) tail: out2 = stack_ptr (8) ++ zeros_like(mem_in) (1024)
// ---------------------------------------------------------------------------
__global__ __launch_bounds__(256) void tail_kernel(const float* __restrict__ sp,
                                                   float* __restrict__ out) {
    int i = blockIdx.x * 256 + threadIdx.x;
    if (i < Ll) out[i] = sp[i];
    else if (i < Ll + 1024) out[i] = 0.f;
}

// ---------------------------------------------------------------------------
extern "C" void kernel_launch(void* const* d_in, const int* in_sizes, int n_in,
                              void* d_out, int out_size, void* d_ws, size_t ws_size,
                              hipStream_t stream) {
    const float* feat       = (const float*)d_in[1];
    const float* feat_edge  = (const float*)d_in[2];
    const float* c_i        = (const float*)d_in[3];
    const float* att_stack  = (const float*)d_in[5];
    const float* stack_ptr  = (const float*)d_in[6];
    const float* mono_mask  = (const float*)d_in[8];
    const float* cross_mask = (const float*)d_in[9];
    const float* w_c        = (const float*)d_in[10];
    const float* b_c        = (const float*)d_in[11];

    float* out = (float*)d_out;
    float* ws  = (float*)d_ws;

    const size_t NN = (size_t)Nn * Nn;      // 1,048,576
    float* query = ws;                      // 64 floats (32 used)
    float* Wv    = ws + 64;                 // N*N
    float* Sxy   = ws + 64 + NN;            // N*N
    float* Cbuf  = ws + 64 + 2 * NN;        // N*N
    float* nrm   = ws + 64 + 3 * NN;        // 1024

    query_kernel<<<1, 32, 0, stream>>>(c_i, w_c, b_c, query);
    wv_kernel<<<dim3(Nn / 256, Nn), 256, 0, stream>>>(query, feat, feat_edge,
                                                      mono_mask, cross_mask, Wv);
    s_kernel<<<(unsigned)(NN / 256), 256, 0, stream>>>(att_stack, stack_ptr, Sxy);
    gemm_kernel<<<dim3(Nn / 64, Nn / 64), 256, 0, stream>>>(Wv, Sxy, Cbuf);
    norm_kernel<<<Nn, 256, 0, stream>>>(Cbuf, nrm);
    final_kernel<<<dim3(Nn / 32, Nn / 32), 256, 0, stream>>>(Cbuf, nrm, att_stack,
                                                             stack_ptr, out);
    tail_kernel<<<5, 256, 0, stream>>>(stack_ptr, out + NN * Ll);
}